// SConv_33294586478860
// MI455X (gfx1250) — compile-verified
//
#include <hip/hip_runtime.h>
#include <cstdint>
#include <cmath>

// Problem sizes (fixed by the reference).
constexpr int BB = 2;
constexpr int LL = 8192;
constexpr int DD = 1024;

// Tiling.
constexpr int DT   = 256;        // channels per block == threads per block
constexpr int NDT  = DD / DT;    // 4 channel tiles
constexpr int TC   = 128;        // timesteps per chunk (scan segment)
constexpr int NC   = LL / TC;    // 64 chunks along time
constexpr int ST   = 32;         // rows per LDS sub-tile (double buffered)
constexpr int NSUB = TC / ST;    // 4 sub-tiles per chunk
// Async b128 segments (16B) per thread per sub-tile.
constexpr int SEG_PER_THREAD = (ST * DT * 4) / 16 / DT;   // == 8
static_assert(SEG_PER_THREAD == 8, "wait-count literal below assumes 8");

// ---------------------------------------------------------------------------
// CDNA5 async global->LDS DMA (ASYNCcnt-tracked). Each lane moves 16 bytes.
// vdst = 32-bit LDS byte address in a VGPR, vaddr = 64-bit global address.
// TH: pass 1 marks x high-temporal (re-read by pass 3 out of L2),
//     pass 3 marks it last-use.
// ---------------------------------------------------------------------------
template <bool KEEP_IN_L2>
__device__ __forceinline__ void issue_subtile_async(const float* gbase,
                                                    unsigned ldsbase, int tid) {
#pragma unroll
  for (int s = 0; s < SEG_PER_THREAD; ++s) {
    const int seg = tid + s * DT;
    const int row = seg >> 6;        // 64 segments (16B) per 256-float row
    const int col = seg & 63;
    const float* gp = gbase + (size_t)row * DD + (col << 2);
    const unsigned lp = ldsbase + (unsigned)(row * (DT * 4) + (col << 4));
    if (KEEP_IN_L2) {
      asm volatile("global_load_async_to_lds_b128 %0, %1, off th:TH_LOAD_HT"
                   :: "v"(lp), "v"(gp) : "memory");
    } else {
      asm volatile("global_load_async_to_lds_b128 %0, %1, off th:TH_LOAD_LU"
                   :: "v"(lp), "v"(gp) : "memory");
    }
  }
}

#define WAIT_ASYNC_ONE_TILE() asm volatile("s_wait_asynccnt 0x8" ::: "memory")
#define WAIT_ASYNC_ALL()      asm volatile("s_wait_asynccnt 0x0" ::: "memory")

// ---------------------------------------------------------------------------
// Pass 1 (WRITE_OUT=false): chunk Horner suffix sum S_c = sum ph^(TC-1-j)*(pinit*x_j)
// Pass 3 (WRITE_OUT=true):  replay g_t = ph*g + pinit*x from chunk start state,
//                           emit Re(g)
// ---------------------------------------------------------------------------
template <bool WRITE_OUT>
__global__ __launch_bounds__(DT) void scan_chunk_kernel(
    const float* __restrict__ x,
    const float* __restrict__ pr_, const float* __restrict__ pi_,
    const float* __restrict__ qr_, const float* __restrict__ qi_,
    float2* __restrict__ S,          // [B][NC][D] (written by pass 1)
    const float2* __restrict__ G,    // [B][NC][D] (read by pass 3)
    float* __restrict__ out)
{
  __shared__ float smem[2][ST * DT];   // 2 x 32KB double buffer

  const int tid  = threadIdx.x;
  const int bid  = blockIdx.x;
  const int c    = bid % NC;
  const int tile = (bid / NC) % NDT;
  const int b    = bid / (NC * NDT);
  const int d    = tile * DT + tid;

  // ph[d] = exp(-|p|^2) * (pr + i*pi)/|p|
  const float pr = pr_[d], pi = pi_[d];
  const float m2 = fmaf(pr, pr, pi * pi);
  const float sc = expf(-m2) * rsqrtf(fmaxf(m2, 1e-30f));
  const float ar = pr * sc, ai = pi * sc;
  const float qr = qr_[d], qi = qi_[d];          // pinit

  const float* xg = x + ((size_t)b * LL + (size_t)c * TC) * DD + tile * DT;

  // Low 32 bits of a generic LDS pointer == LDS byte offset (ISA 10.2).
  const unsigned lds0 = (unsigned)(uintptr_t)&smem[0][0];
  const unsigned lds1 = (unsigned)(uintptr_t)&smem[1][0];

  float gr, gi;
  if (WRITE_OUT) {
    const float2 g0 = G[((size_t)b * NC + c) * DD + d];
    gr = g0.x; gi = g0.y;
  } else {
    gr = 0.0f; gi = 0.0f;
  }
  float* op = WRITE_OUT ? (out + ((size_t)b * LL + (size_t)c * TC) * DD + d)
                        : nullptr;

  issue_subtile_async<!WRITE_OUT>(xg, lds0, tid);

  for (int sub = 0; sub < NSUB; ++sub) {
    if (sub + 1 < NSUB) {
      // Prefetch next sub-tile into the other buffer, then wait until only
      // those 8 ops remain outstanding -> current sub-tile is resident.
      issue_subtile_async<!WRITE_OUT>(xg + (size_t)(sub + 1) * ST * DD,
                                      (sub & 1) ? lds0 : lds1, tid);
      WAIT_ASYNC_ONE_TILE();
    } else {
      WAIT_ASYNC_ALL();
    }
    __syncthreads();

    const float* buf = (sub & 1) ? &smem[1][0] : &smem[0][0];
#pragma unroll
    for (int j = 0; j < ST; ++j) {
      const float xv  = buf[j * DT + tid];                 // conflict-free ds_load
      const float ngr = fmaf(ar, gr, fmaf(-ai, gi, qr * xv));
      const float ngi = fmaf(ar, gi, fmaf( ai, gr, qi * xv));
      gr = ngr; gi = ngi;
      if (WRITE_OUT) op[(size_t)(sub * ST + j) * DD] = gr; // coalesced 1KB rows
    }
    __syncthreads();   // everyone done reading before this buffer is re-filled
  }

  if (!WRITE_OUT) {
    S[((size_t)b * NC + c) * DD + d] = make_float2(gr, gi);
  }
}

// ---------------------------------------------------------------------------
// Pass 2: serial scan over chunk states per (b,d):
//   G_0 = last;  G_c = ph^TC * G_{c-1} + S_{c-1}
// ph^TC computed by 7 complex squarings (TC = 128 = 2^7): branch-free, and
// rounding-consistent with the pass-1 Horner chain.
// ---------------------------------------------------------------------------
__global__ __launch_bounds__(256) void combine_kernel(
    const float* __restrict__ pr_, const float* __restrict__ pi_,
    const float* __restrict__ lr_, const float* __restrict__ li_,
    const float2* __restrict__ S, float2* __restrict__ G)
{
  const int idx = blockIdx.x * blockDim.x + threadIdx.x;
  if (idx >= BB * DD) return;
  const int b = idx / DD, d = idx % DD;

  const float pr = pr_[d], pi = pi_[d];
  const float m2 = fmaf(pr, pr, pi * pi);
  const float sc = expf(-m2) * rsqrtf(fmaxf(m2, 1e-30f));
  float Ar = pr * sc, Ai = pi * sc;            // ph
#pragma unroll
  for (int k = 0; k < 7; ++k) {                // ph^(2^7) = ph^TC
    const float nr = fmaf(Ar, Ar, -(Ai * Ai));
    const float ni = 2.0f * Ar * Ai;
    Ar = nr; Ai = ni;
  }

  float gr = lr_[d], gi = li_[d];
  G[((size_t)b * NC) * DD + d] = make_float2(gr, gi);
  for (int c = 1; c < NC; ++c) {
    const float2 s = S[((size_t)b * NC + (c - 1)) * DD + d];  // coalesced over d
    const float ngr = fmaf(Ar, gr, fmaf(-Ai, gi, s.x));
    const float ngi = fmaf(Ar, gi, fmaf( Ai, gr, s.y));
    gr = ngr; gi = ngi;
    G[((size_t)b * NC + c) * DD + d] = make_float2(gr, gi);
  }
}

extern "C" void kernel_launch(void* const* d_in, const int* in_sizes, int n_in,
                              void* d_out, int out_size, void* d_ws, size_t ws_size,
                              hipStream_t stream) {
  const float* x  = (const float*)d_in[0];
  const float* pr = (const float*)d_in[1];
  const float* pi = (const float*)d_in[2];
  const float* qr = (const float*)d_in[3];
  const float* qi = (const float*)d_in[4];
  const float* lr = (const float*)d_in[5];
  const float* li = (const float*)d_in[6];
  float* out = (float*)d_out;

  float2* S = (float2*)d_ws;                          // 1 MiB
  float2* G = S + (size_t)BB * NC * DD;               // 1 MiB

  const dim3 blk(DT);
  const dim3 grd(BB * NDT * NC);                      // 512 blocks, 8 waves each

  scan_chunk_kernel<false><<<grd, blk, 0, stream>>>(x, pr, pi, qr, qi,
                                                    S, nullptr, nullptr);
  combine_kernel<<<(BB * DD + 255) / 256, 256, 0, stream>>>(pr, pi, lr, li, S, G);
  scan_chunk_kernel<true><<<grd, blk, 0, stream>>>(x, pr, pi, qr, qi,
                                                   nullptr, G, out);
}